// mLSTM_33071248179633
// MI455X (gfx1250) — compile-verified
//
#include <hip/hip_runtime.h>
#include <math.h>

#define BDIM 256
#define DDIM 1024
#define HHEADS 8
#define DH 128
#define PDIM 2048
#define KCONV 4
#define HD 1024
#define EPSF 1e-5f

typedef float v2f __attribute__((ext_vector_type(2)));
typedef float v4f __attribute__((ext_vector_type(4)));
typedef float v8f __attribute__((ext_vector_type(8)));

enum { EPI_BIAS = 0, EPI_BIAS_SCALE = 1, EPI_SIGMOID = 2, EPI_NOBIAS = 3, EPI_BIAS_RES = 4 };

// ---------------------------------------------------------------------------
// fp32 WMMA GEMM: C[M,N] = epilogue(A[M,K] @ W[N,K]^T + bias)
// One wave computes a 32x32 output block as 4 x (16x16) WMMA tiles:
// per k-step of 4 we issue 4 contiguous float2 loads (2 A-frags, 2 B-frags)
// and 4 v_wmma_f32_16x16x4_f32 -> 8 flop/byte from L2 (2x the 1-tile version).
// Fragment layout (ISA 7.12.2, 32-bit 16x4 A / 4x16 B):
//   lane l<16  -> row (base+l), K={k,k+1};  lane l>=16 -> row (base+l-16), K={k+2,k+3}
// D layout: VGPR i, lanes 0-15 -> M=i, lanes 16-31 -> M=8+i; N=lane&15.
// ---------------------------------------------------------------------------
template <int EPI>
__global__ __launch_bounds__(32) void gemm_nt_wmma(
    const float* __restrict__ A, const float* __restrict__ W,
    const float* __restrict__ bias, const float* __restrict__ res,
    float* __restrict__ C, int M, int N, int Kd, float scale) {
  const int lane = threadIdx.x;
  const int half = lane >> 4;
  const int l = lane & 15;
  const int n0 = blockIdx.x * 32;
  const int m0 = blockIdx.y * 32;

  const float* ap0 = A + (size_t)(m0 + l) * Kd + 2 * half;
  const float* ap1 = ap0 + (size_t)16 * Kd;
  const float* wp0 = W + (size_t)(n0 + l) * Kd + 2 * half;
  const float* wp1 = wp0 + (size_t)16 * Kd;

  v8f acc00 = {}, acc01 = {}, acc10 = {}, acc11 = {};
#pragma unroll 2
  for (int k = 0; k < Kd; k += 4) {
    v2f a0 = *(const v2f*)(ap0 + k);
    v2f a1 = *(const v2f*)(ap1 + k);
    v2f b0 = *(const v2f*)(wp0 + k);
    v2f b1 = *(const v2f*)(wp1 + k);
    acc00 = __builtin_amdgcn_wmma_f32_16x16x4_f32(false, a0, false, b0,
                                                  (short)0, acc00, false, false);
    acc01 = __builtin_amdgcn_wmma_f32_16x16x4_f32(false, a0, false, b1,
                                                  (short)0, acc01, false, false);
    acc10 = __builtin_amdgcn_wmma_f32_16x16x4_f32(false, a1, false, b0,
                                                  (short)0, acc10, false, false);
    acc11 = __builtin_amdgcn_wmma_f32_16x16x4_f32(false, a1, false, b1,
                                                  (short)0, acc11, false, false);
  }

  auto store_tile = [&](int mb, int nb, v8f acc) {
    const int col = nb + l;
    const float bsv = (EPI == EPI_NOBIAS) ? 0.f : bias[col];
#pragma unroll
    for (int i = 0; i < 8; ++i) {
      const int row = mb + half * 8 + i;
      float x = acc[i] + bsv;
      if (EPI == EPI_BIAS_SCALE) x *= scale;
      if (EPI == EPI_SIGMOID) x = 1.f / (1.f + expf(-x));
      if (EPI == EPI_BIAS_RES) x += res[(size_t)row * N + col];
      C[(size_t)row * N + col] = x;
    }
  };
  store_tile(m0, n0, acc00);
  store_tile(m0, n0 + 16, acc01);
  store_tile(m0 + 16, n0, acc10);
  store_tile(m0 + 16, n0 + 16, acc11);
}

// ---------------------------------------------------------------------------
// LayerNorm over D=1024, one block (256 threads) per batch row.
// ---------------------------------------------------------------------------
__global__ __launch_bounds__(256) void ln_kernel(const float* __restrict__ x,
                                                 const float* __restrict__ g,
                                                 const float* __restrict__ bvec,
                                                 float* __restrict__ y) {
  __shared__ float red[256];
  const int b = blockIdx.x, t = threadIdx.x;
  const float* xr = x + (size_t)b * DDIM;
  v4f xv = *(const v4f*)(xr + t * 4);
  red[t] = xv.x + xv.y + xv.z + xv.w;
  __syncthreads();
  for (int s = 128; s > 0; s >>= 1) { if (t < s) red[t] += red[t + s]; __syncthreads(); }
  const float mu = red[0] / (float)DDIM;
  __syncthreads();
  v4f dx = xv - mu;
  red[t] = dx.x * dx.x + dx.y * dx.y + dx.z * dx.z + dx.w * dx.w;
  __syncthreads();
  for (int s = 128; s > 0; s >>= 1) { if (t < s) red[t] += red[t + s]; __syncthreads(); }
  const float rstd = rsqrtf(red[0] / (float)DDIM + EPSF);
  v4f gv = *(const v4f*)(g + t * 4);
  v4f bv = *(const v4f*)(bvec + t * 4);
  v4f yv = dx * rstd * gv + bv;
  *(v4f*)(y + (size_t)b * DDIM + t * 4) = yv;
}

// ---------------------------------------------------------------------------
// Causal conv1d (K=4, left pad 3) along P, then SiLU.
// ---------------------------------------------------------------------------
__global__ __launch_bounds__(256) void conv_silu_kernel(const float* __restrict__ lt,
                                                        const float* __restrict__ cw,
                                                        const float* __restrict__ cb,
                                                        float* __restrict__ lc) {
  const int idx = blockIdx.x * 256 + threadIdx.x;
  const int b = idx >> 11;           // / PDIM
  const int p = idx & (PDIM - 1);
  float acc = cb[0];
#pragma unroll
  for (int k = 0; k < KCONV; ++k) {
    const int src = p + k - (KCONV - 1);
    if (src >= 0) acc += cw[k] * lt[(size_t)b * PDIM + src];
  }
  lc[idx] = acc / (1.f + expf(-acc));  // silu
}

// ---------------------------------------------------------------------------
// i/f gates: it = lc @ Wi^T + bi, ft = lc @ Wf^T + bf   (N=8, tiny GEMV)
// ---------------------------------------------------------------------------
__global__ __launch_bounds__(256) void ifgate_kernel(const float* __restrict__ lc,
                                                     const float* __restrict__ Wi,
                                                     const float* __restrict__ bi,
                                                     const float* __restrict__ Wf,
                                                     const float* __restrict__ bf,
                                                     float* __restrict__ itv,
                                                     float* __restrict__ ftv) {
  __shared__ float red[256];
  const int b = blockIdx.x, t = threadIdx.x;
  float ai[HHEADS] = {}, af[HHEADS] = {};
  for (int p = t; p < PDIM; p += 256) {
    const float x = lc[(size_t)b * PDIM + p];
#pragma unroll
    for (int h = 0; h < HHEADS; ++h) {
      ai[h] += x * Wi[h * PDIM + p];
      af[h] += x * Wf[h * PDIM + p];
    }
  }
  for (int h = 0; h < HHEADS; ++h) {
    red[t] = ai[h]; __syncthreads();
    for (int s = 128; s > 0; s >>= 1) { if (t < s) red[t] += red[t + s]; __syncthreads(); }
    if (t == 0) itv[b * HHEADS + h] = red[0] + bi[h];
    __syncthreads();
    red[t] = af[h]; __syncthreads();
    for (int s = 128; s > 0; s >>= 1) { if (t < s) red[t] += red[t + s]; __syncthreads(); }
    if (t == 0) ftv[b * HHEADS + h] = red[0] + bf[h];
    __syncthreads();
  }
}

// ---------------------------------------------------------------------------
// Fused state update, one block per (b,h):
//   mt/i_/f_, nt + nq reduction, ct stream (NT load/store) fused with ct@q,
//   htt/ht, per-head GroupNorm, + skip, * silu(rt) -> y.
// ---------------------------------------------------------------------------
__global__ __launch_bounds__(256) void state_kernel(
    const float* __restrict__ c_prev, const float* __restrict__ n_prev,
    const float* __restrict__ m_prev, const float* __restrict__ qt,
    const float* __restrict__ kt, const float* __restrict__ vt,
    const float* __restrict__ ot, const float* __restrict__ itv,
    const float* __restrict__ ftv, const float* __restrict__ skip,
    const float* __restrict__ rt, const float* __restrict__ gn_g,
    const float* __restrict__ gn_b, float* __restrict__ ct_out,
    float* __restrict__ nt_out, float* __restrict__ mt_out,
    float* __restrict__ y) {
  __shared__ __align__(16) float sq[DH];
  __shared__ __align__(16) float sk[DH];
  __shared__ __align__(16) float sv[DH];
  __shared__ float red[256];
  __shared__ float cq[DH];
  __shared__ float s_nqt;

  const int bh = blockIdx.x;
  const int h = bh & (HHEADS - 1);
  const int t = threadIdx.x;
  const size_t base128 = (size_t)bh * DH;          // == b*HD + h*DH
  const size_t baseC = (size_t)bh * DH * DH;

  const float it_ = itv[bh], ft_ = ftv[bh], mp = m_prev[bh];
  const float mt = fmaxf(ft_ + mp, it_);
  const float i_ = expf(it_ - mt);
  const float f_ = expf(ft_ - mt + mp);
  if (t == 0) mt_out[bh] = mt;

  if (t < DH) {
    sq[t] = qt[base128 + t];
    sk[t] = kt[base128 + t];
    sv[t] = vt[base128 + t];
  }
  __syncthreads();

  // nt and nq = dot(nt, q)
  float part = 0.f;
  if (t < DH) {
    const float ntv = f_ * n_prev[base128 + t] + i_ * sk[t];
    nt_out[base128 + t] = ntv;
    part = ntv * sq[t];
  }
  red[t] = part; __syncthreads();
  for (int s = 128; s > 0; s >>= 1) { if (t < s) red[t] += red[t + s]; __syncthreads(); }
  if (t == 0) s_nqt = fminf(red[0], 1.f);
  __syncthreads();
  const float nqt = s_nqt;

  // ct = f_*c_prev + i_*v k^T  (non-temporal 256 MB stream), fused cq[d]=ct@q
  const int d = t >> 1;
  const int ebase = (t & 1) * 64;
  const float* cp = c_prev + baseC + (size_t)d * DH + ebase;
  float* co = ct_out + baseC + (size_t)d * DH + ebase;
  const float iv = i_ * sv[d];
  float qacc = 0.f;
#pragma unroll
  for (int e = 0; e < 64; e += 4) {
    v4f cpv = __builtin_nontemporal_load((const v4f*)(cp + e));
    v4f kv = *(const v4f*)&sk[ebase + e];
    v4f qv = *(const v4f*)&sq[ebase + e];
    v4f cv = f_ * cpv + iv * kv;
    __builtin_nontemporal_store(cv, (v4f*)(co + e));
    qacc += cv.x * qv.x + cv.y * qv.y + cv.z * qv.z + cv.w * qv.w;
  }
  __syncthreads();
  red[t] = qacc; __syncthreads();
  if ((t & 1) == 0) cq[t >> 1] = red[t] + red[t + 1];
  __syncthreads();

  // ht, per-head GroupNorm, skip add, silu(rt) gate
  float ht = 0.f;
  if (t < DH) ht = ot[base128 + t] * (cq[t] / nqt);
  red[t] = (t < DH) ? ht : 0.f; __syncthreads();
  for (int s = 128; s > 0; s >>= 1) { if (t < s) red[t] += red[t + s]; __syncthreads(); }
  const float gm = red[0] / (float)DH;
  __syncthreads();
  const float dv = (t < DH) ? (ht - gm) : 0.f;
  red[t] = dv * dv; __syncthreads();
  for (int s = 128; s > 0; s >>= 1) { if (t < s) red[t] += red[t + s]; __syncthreads(); }
  const float gvv = red[0] / (float)DH;
  __syncthreads();
  if (t < DH) {
    const int ch = h * DH + t;
    float gn = (ht - gm) * rsqrtf(gvv + EPSF) * gn_g[ch] + gn_b[ch];
    float o1 = gn + skip[base128 + t];
    const float r = rt[base128 + t];
    o1 *= r / (1.f + expf(-r));
    y[base128 + t] = o1;
  }
}

// ---------------------------------------------------------------------------
extern "C" void kernel_launch(void* const* d_in, const int* in_sizes, int n_in,
                              void* d_out, int out_size, void* d_ws, size_t ws_size,
                              hipStream_t stream) {
  const float* sequence = (const float*)d_in[0];
  const float* c_prev   = (const float*)d_in[1];
  const float* n_prev   = (const float*)d_in[2];
  const float* m_prev   = (const float*)d_in[3];
  const float* ln_g     = (const float*)d_in[4];
  const float* ln_b     = (const float*)d_in[5];
  const float* Wl       = (const float*)d_in[6];
  const float* bl       = (const float*)d_in[7];
  const float* Wr       = (const float*)d_in[8];
  const float* br       = (const float*)d_in[9];
  const float* Wd       = (const float*)d_in[10];
  const float* bd       = (const float*)d_in[11];
  const float* conv_w   = (const float*)d_in[12];
  const float* conv_b   = (const float*)d_in[13];
  const float* Wskip    = (const float*)d_in[14];
  const float* Wi       = (const float*)d_in[15];
  const float* bi       = (const float*)d_in[16];
  const float* Wf       = (const float*)d_in[17];
  const float* bf       = (const float*)d_in[18];
  const float* Wo       = (const float*)d_in[19];
  const float* bo       = (const float*)d_in[20];
  const float* Wq       = (const float*)d_in[21];
  const float* bq       = (const float*)d_in[22];
  const float* Wk       = (const float*)d_in[23];
  const float* bk       = (const float*)d_in[24];
  const float* Wv       = (const float*)d_in[25];
  const float* bv       = (const float*)d_in[26];
  const float* gn_g     = (const float*)d_in[27];
  const float* gn_b     = (const float*)d_in[28];

  float* out    = (float*)d_out;                                   // [B,D]
  float* ct_out = out + (size_t)BDIM * DDIM;                       // [B,H,Dh,Dh]
  float* nt_out = ct_out + (size_t)BDIM * HHEADS * DH * DH;        // [B,H,Dh]
  float* mt_out = nt_out + (size_t)BDIM * HHEADS * DH;             // [B,H]

  float* ws   = (float*)d_ws;
  float* xn   = ws;  ws += (size_t)BDIM * DDIM;
  float* lt   = ws;  ws += (size_t)BDIM * PDIM;
  float* rt   = ws;  ws += (size_t)BDIM * HD;
  float* lc   = ws;  ws += (size_t)BDIM * PDIM;
  float* qt   = ws;  ws += (size_t)BDIM * HD;
  float* kt   = ws;  ws += (size_t)BDIM * HD;
  float* vt   = ws;  ws += (size_t)BDIM * HD;
  float* ot   = ws;  ws += (size_t)BDIM * HD;
  float* skp  = ws;  ws += (size_t)BDIM * HHEADS * DH;
  float* itv  = ws;  ws += (size_t)BDIM * HHEADS;
  float* ftv  = ws;  ws += (size_t)BDIM * HHEADS;
  float* yact = ws;  ws += (size_t)BDIM * HD;

  const float kscale = 0.08838834764831845f;  // 1/sqrt(128)

  ln_kernel<<<BDIM, 256, 0, stream>>>(sequence, ln_g, ln_b, xn);

  gemm_nt_wmma<EPI_BIAS><<<dim3(PDIM / 32, BDIM / 32), 32, 0, stream>>>(
      xn, Wl, bl, nullptr, lt, BDIM, PDIM, DDIM, 1.f);
  gemm_nt_wmma<EPI_BIAS><<<dim3(HD / 32, BDIM / 32), 32, 0, stream>>>(
      xn, Wr, br, nullptr, rt, BDIM, HD, DDIM, 1.f);

  conv_silu_kernel<<<(BDIM * PDIM) / 256, 256, 0, stream>>>(lt, conv_w, conv_b, lc);

  gemm_nt_wmma<EPI_BIAS><<<dim3(HD / 32, BDIM / 32), 32, 0, stream>>>(
      lc, Wq, bq, nullptr, qt, BDIM, HD, PDIM, 1.f);
  gemm_nt_wmma<EPI_BIAS_SCALE><<<dim3(HD / 32, BDIM / 32), 32, 0, stream>>>(
      lc, Wk, bk, nullptr, kt, BDIM, HD, PDIM, kscale);
  gemm_nt_wmma<EPI_BIAS><<<dim3(HD / 32, BDIM / 32), 32, 0, stream>>>(
      lt, Wv, bv, nullptr, vt, BDIM, HD, PDIM, 1.f);
  gemm_nt_wmma<EPI_SIGMOID><<<dim3(HD / 32, BDIM / 32), 32, 0, stream>>>(
      lt, Wo, bo, nullptr, ot, BDIM, HD, PDIM, 1.f);
  gemm_nt_wmma<EPI_NOBIAS><<<dim3(HD / 32, BDIM / 32), 32, 0, stream>>>(
      lc, Wskip, nullptr, nullptr, skp, BDIM, HD, PDIM, 1.f);

  ifgate_kernel<<<BDIM, 256, 0, stream>>>(lc, Wi, bi, Wf, bf, itv, ftv);

  state_kernel<<<BDIM * HHEADS, 256, 0, stream>>>(
      c_prev, n_prev, m_prev, qt, kt, vt, ot, itv, ftv, skp, rt, gn_g, gn_b,
      ct_out, nt_out, mt_out, yact);

  gemm_nt_wmma<EPI_BIAS_RES><<<dim3(DDIM / 32, BDIM / 32), 32, 0, stream>>>(
      yact, Wd, bd, sequence, out, BDIM, DDIM, HD, 1.f);
}